// CluttrVAE_89326729822553
// MI455X (gfx1250) — compile-verified
//
#include <hip/hip_runtime.h>
#include <cstdint>
#include <cstddef>

// ---------------------------------------------------------------------------
// CluttrVAE forward on MI455X (gfx1250), fp32 throughout.
// Matrix path: V_WMMA_F32_16X16X4_F32 (native CDNA5 f32 WMMA, f32 accum).
// GEMM: LDS-staged, double-buffered K-chunks; coalesced global loads;
// bank-conflict-free padded LDS (stride 34 dwords); branch-free edge
// handling via address clamping; template-specialized epilogues (no runtime
// branches); unrolled full-chunk inner loop (8x[4 ds_load_b64 + 4 wmma]);
// dual A/B slot fuses x_t@Wi + h@Wh for LSTM gates.
// Workspace requirement: ~258 MB.
// ---------------------------------------------------------------------------

typedef float v2f __attribute__((ext_vector_type(2)));
typedef float v8f __attribute__((ext_vector_type(8)));

#define BB     512
#define SEQL   52
#define NVOCAB 170
#define DEMB   300
#define DLAT   64
#define HENC   300
#define HDEC   400
#define NTOK   (BB * SEQL)   // 26624

static __device__ __forceinline__ float sigf(float v) {
  return 1.0f / (1.0f + __expf(-v));
}
static __device__ __forceinline__ int imin(int a, int b) { return a < b ? a : b; }

static __device__ __forceinline__ v8f wmma4(v2f a, v2f b, v8f c) {
  // D = A(16x4,f32) * B(4x16,f32) + C(16x16,f32)
  return __builtin_amdgcn_wmma_f32_16x16x4_f32(
      /*neg_a=*/false, a, /*neg_b=*/false, b,
      /*c_mod=*/(short)0, c, /*reuse_a=*/false, /*reuse_b=*/false);
}

// ---------------------------------------------------------------------------
// Generic GEMM:  D = act( A1*B1 [+ A2*B2] + bias + Cin )
//   block = 128 threads = 4 waves; block tile 64x64; wave tile 32x32
//   (2x2 WMMA fragments).  M must be a multiple of 64 (true: 26624, 512).
//   K must be a multiple of 4 (true: 300/600/64/400/800/1600).  N masked.
// ACT: 0 none | 1 relu | 3 4*tanh | 4 highway-combine
//   ACT 4:  g = sigmoid(acc+bias); D = g*T2 + (1-g)*T4
// ---------------------------------------------------------------------------
template <int ACT, bool DUAL, bool HAS_CIN, bool HAS_BIAS>
__global__ __launch_bounds__(128) void k_gemm(
    const float* __restrict__ A1, int lda1, const float* __restrict__ B1, int ldb1, int K1,
    const float* __restrict__ A2, int lda2, const float* __restrict__ B2, int ldb2, int K2,
    const float* __restrict__ bias, const float* __restrict__ Cin,
    const float* __restrict__ T2, const float* __restrict__ T4,
    float* __restrict__ D, int ldd, int N)
{
  constexpr int KC = 32;    // K-chunk staged in LDS
  constexpr int LS = 34;    // padded LDS row stride (dwords): conflict-free

  // A chunk: 64 rows x KC cols, row-major, stride LS.
  // B chunk: KC rows x 64 cols, stored TRANSPOSED (col-major, stride LS) so a
  // B fragment (two consecutive K rows, one column) is a single ds_load_b64.
  __shared__ float As[2][64 * LS];
  __shared__ float Bs[2][64 * LS];

  const int tid  = threadIdx.x;
  const int lane = tid & 31;
  const int wid  = tid >> 5;
  const int r    = lane & 15;
  const int half = lane >> 4;
  const int wm   = wid >> 1, wn = wid & 1;
  const int tm0  = blockIdx.y * 64;
  const int tn0  = blockIdx.x * 64;

  v8f acc[2][2];
  const v8f vzero = {0.f, 0.f, 0.f, 0.f, 0.f, 0.f, 0.f, 0.f};
  acc[0][0] = vzero; acc[0][1] = vzero; acc[1][0] = vzero; acc[1][1] = vzero;

  // cooperative-load decomposition (per thread, 16 elements each):
  //   A: col = tid&31 (fixed), rows (tid>>5) + 4*i   -> coalesced 128B rows
  //   B: col = tid&63 (fixed), rows (tid>>6) + 2*i   -> coalesced 256B rows
  const int acol0 = tid & 31;
  const int arow0 = tid >> 5;
  const int bcol0 = tid & 63;
  const int brow0 = tid >> 6;

  float ra[16], rb[16];

  const float* A  = A1;  const float* Bm = B1;
  int lda = lda1, ldb = ldb1, K = K1;

#pragma unroll
  for (int slot = 0; slot < (DUAL ? 2 : 1); ++slot) {
    if (slot == 1) {
      A = A2; Bm = B2; lda = lda2; ldb = ldb2; K = K2;
      __syncthreads();   // slot-1 LDS stores must not race slot-0 reads
    }

    // Branch-free guarded global loads: clamp OOB addresses (values beyond
    // K are never read by the inner loop; beyond-N columns are masked at the
    // epilogue store), so no exec-mask manipulation is ever needed.
    const int bc = imin(tn0 + bcol0, N - 1);           // loop-invariant

    auto load_chunk = [&](int k0) {
      const int ac = imin(k0 + acol0, K - 1);
      const float* Ap = A + (tm0 + arow0) * lda + ac;
#pragma unroll
      for (int i = 0; i < 16; ++i) ra[i] = Ap[(4 * i) * lda];
#pragma unroll
      for (int i = 0; i < 16; ++i) {
        const int br = imin(k0 + brow0 + 2 * i, K - 1);
        rb[i] = Bm[br * ldb + bc];
      }
    };

    const int nch = (K + KC - 1) / KC;
    load_chunk(0);

    for (int ci = 0; ci < nch; ++ci) {
      const int buf = ci & 1;
      // stage registers -> LDS (conflict-free: A rows hit consecutive banks,
      // B transposed stores stride 34 -> 17*c mod 32 is a permutation)
#pragma unroll
      for (int i = 0; i < 16; ++i) {
        As[buf][(arow0 + 4 * i) * LS + acol0] = ra[i];
        Bs[buf][bcol0 * LS + brow0 + 2 * i]  = rb[i];
      }
      __syncthreads();

      if (ci + 1 < nch) load_chunk((ci + 1) * KC);  // overlap with compute

      const int kcLen = imin(K - ci * KC, KC);

      const float* as0 = &As[buf][(wm * 32 + r) * LS + 2 * half];
      const float* as1 = as0 + 16 * LS;
      const float* bs0 = &Bs[buf][(wn * 32 + r) * LS + 2 * half];
      const float* bs1 = bs0 + 16 * LS;

      auto step = [&](int s4) {
        v2f a0 = *(const v2f*)(as0 + s4);
        v2f a1 = *(const v2f*)(as1 + s4);
        v2f b0 = *(const v2f*)(bs0 + s4);
        v2f b1 = *(const v2f*)(bs1 + s4);
        acc[0][0] = wmma4(a0, b0, acc[0][0]);
        acc[0][1] = wmma4(a0, b1, acc[0][1]);
        acc[1][0] = wmma4(a1, b0, acc[1][0]);
        acc[1][1] = wmma4(a1, b1, acc[1][1]);
      };

      if (kcLen == KC) {
        // hot path: compile-time trip count -> full unroll, loads batched
#pragma unroll
        for (int s4 = 0; s4 < KC; s4 += 4) step(s4);
      } else {
        for (int s4 = 0; s4 < kcLen; s4 += 4) step(s4);
      }
    }
  }

  // Epilogue.  C/D layout (ISA 7.12.2): lane l, VGPR i -> row i + 8*(l>>4),
  // col l&15 within each 16x16 tile.  All offsets fit in 32 bits.
#pragma unroll
  for (int im = 0; im < 2; ++im) {
#pragma unroll
    for (int in = 0; in < 2; ++in) {
      const int col = tn0 + wn * 32 + in * 16 + r;
      if (col >= N) continue;
      float bv = 0.f;
      if constexpr (HAS_BIAS) bv = bias[col];
      const int base = (tm0 + wm * 32 + im * 16 + 8 * half) * ldd + col;
#pragma unroll
      for (int i = 0; i < 8; ++i) {
        const int off = base + i * ldd;
        float v = acc[im][in][i] + bv;
        if constexpr (HAS_CIN) v += Cin[off];
        if constexpr (ACT == 1) v = fmaxf(v, 0.f);
        else if constexpr (ACT == 3) v = 4.0f * tanhf(v);
        else if constexpr (ACT == 4) {
          float g = sigf(v);
          v = g * T2[off] + (1.f - g) * T4[off];
        }
        D[off] = v;
      }
    }
  }
}

// ---------------------------------------------------------------------------
// Elementwise kernels
// ---------------------------------------------------------------------------
__global__ __launch_bounds__(256) void k_fill0(float* p, int n) {
  int i = blockIdx.x * blockDim.x + threadIdx.x;
  if (i < n) p[i] = 0.f;
}

__global__ __launch_bounds__(256) void k_embed(const int* __restrict__ x,
                                               const float* __restrict__ emb,
                                               float* __restrict__ e, int n) {
  int i = blockIdx.x * blockDim.x + threadIdx.x;
  if (i >= n) return;
  int tok = i / DEMB;
  int j   = i - tok * DEMB;
  e[i] = emb[x[tok] * DEMB + j];
}

// LSTM cell: gates (BB x 4H) laid out [i|f|g|o]; updates c,h; writes h into
// the (B,SEQ,*) sequence buffer at time t_out with column offset colOff.
__global__ __launch_bounds__(256) void k_cell(const float* __restrict__ gates,
                                              float* __restrict__ c, float* __restrict__ h,
                                              int H, float* __restrict__ outSeq,
                                              int ldout, int colOff, int t_out) {
  int i = blockIdx.x * blockDim.x + threadIdx.x;
  if (i >= BB * H) return;
  int b = i / H, j = i - b * H;
  const float* g4 = gates + b * 4 * H;
  float ig = sigf(g4[j]);
  float fg = sigf(g4[H + j]);
  float gg = tanhf(g4[2 * H + j]);
  float og = sigf(g4[3 * H + j]);
  float cn = fg * c[i] + ig * gg;
  float hn = og * tanhf(cn);
  c[i] = cn;
  h[i] = hn;
  outSeq[(b * SEQL + t_out) * ldout + colOff + j] = hn;
}

// hcat = concat(enc[:, -1, :HENC], enc[:, 0, HENC:])
__global__ __launch_bounds__(256) void k_hcat(const float* __restrict__ enc,
                                              float* __restrict__ hcat) {
  int i = blockIdx.x * blockDim.x + threadIdx.x;
  if (i >= BB * 2 * HENC) return;
  int b = i / (2 * HENC), j = i - b * 2 * HENC;
  int t = (j < HENC) ? (SEQL - 1) : 0;
  hcat[i] = enc[(b * SEQL + t) * (2 * HENC) + j];
}

// z = mean + eps * exp(0.5*logvar); write to d_out z region and ws copy
__global__ __launch_bounds__(256) void k_z(const float* __restrict__ mean,
                                           const float* __restrict__ logvar,
                                           const float* __restrict__ eps,
                                           float* __restrict__ z_out,
                                           float* __restrict__ z_ws) {
  int i = blockIdx.x * blockDim.x + threadIdx.x;
  if (i >= BB * DLAT) return;
  float zv = mean[i] + eps[i] * __expf(0.5f * logvar[i]);
  z_out[i] = zv;
  z_ws[i]  = zv;
}

// ---------------------------------------------------------------------------
// Host-side helpers
// ---------------------------------------------------------------------------
template <int ACT, bool DUAL, bool HAS_CIN, bool HAS_BIAS>
static void gemm(hipStream_t st,
                 const float* A1, int lda1, const float* B1, int ldb1, int K1,
                 const float* A2, int lda2, const float* B2, int ldb2, int K2,
                 const float* bias, const float* Cin, const float* T2, const float* T4,
                 float* D, int ldd, int M, int N) {
  dim3 grid((unsigned)((N + 63) / 64), (unsigned)(M / 64));
  k_gemm<ACT, DUAL, HAS_CIN, HAS_BIAS><<<grid, 128, 0, st>>>(
      A1, lda1, B1, ldb1, K1, A2, lda2, B2, ldb2, K2,
      bias, Cin, T2, T4, D, ldd, N);
}

static void fill0(hipStream_t st, float* p, int n) {
  k_fill0<<<(n + 255) / 256, 256, 0, st>>>(p, n);
}

extern "C" void kernel_launch(void* const* d_in, const int* in_sizes, int n_in,
                              void* d_out, int out_size, void* d_ws, size_t ws_size,
                              hipStream_t stream) {
  (void)in_sizes; (void)n_in; (void)out_size; (void)ws_size;

  // ---- inputs (setup_inputs order) ----
  const int*   x     = (const int*)d_in[0];
  const float* eps   = (const float*)d_in[1];
  const float* emb   = (const float*)d_in[2];
  const float* hw0w[5], *hw0b[5], *hw1w[5], *hw1b[5];
  for (int i = 0; i < 5; ++i) { hw0w[i] = (const float*)d_in[3 + 2 * i];  hw0b[i] = (const float*)d_in[4 + 2 * i]; }
  for (int i = 0; i < 5; ++i) { hw1w[i] = (const float*)d_in[13 + 2 * i]; hw1b[i] = (const float*)d_in[14 + 2 * i]; }
  const float* enc_wi[2] = { (const float*)d_in[23], (const float*)d_in[26] };
  const float* enc_wh[2] = { (const float*)d_in[24], (const float*)d_in[27] };
  const float* enc_bb[2] = { (const float*)d_in[25], (const float*)d_in[28] };
  const float* mean_w  = (const float*)d_in[29];
  const float* mean_b  = (const float*)d_in[30];
  const float* logv_w  = (const float*)d_in[31];
  const float* logv_b  = (const float*)d_in[32];
  const float* d1_wi[2] = { (const float*)d_in[33], (const float*)d_in[36] };
  const float* d1_wh[2] = { (const float*)d_in[34], (const float*)d_in[37] };
  const float* d1_bb[2] = { (const float*)d_in[35], (const float*)d_in[38] };
  const float* d2_wi[2] = { (const float*)d_in[39], (const float*)d_in[42] };
  const float* d2_wh[2] = { (const float*)d_in[40], (const float*)d_in[43] };
  const float* d2_bb[2] = { (const float*)d_in[41], (const float*)d_in[44] };
  const float* out_w   = (const float*)d_in[45];
  const float* out_b   = (const float*)d_in[46];

  // ---- outputs: logits | mean | logvar | z (flat) ----
  float* out_logits = (float*)d_out;
  float* out_mean   = out_logits + (size_t)NTOK * NVOCAB;   // 4,526,080
  float* out_logvar = out_mean   + (size_t)BB * DLAT;
  float* out_z      = out_logvar + (size_t)BB * DLAT;

  // ---- workspace arena (phase-reuse; ~258 MB total) ----
  float* W = (float*)d_ws;
  const size_t P = (size_t)NTOK * DEMB;             // 7,987,200 floats
  float* R0 = W;          float* R1 = W + P;  float* R2 = W + 2 * P;
  float* R3 = W + 3 * P;  float* R4 = W + 4 * P;
  float* ex = W + 5 * P;
  float* gates = ex;                 ex += (size_t)BB * 4 * HDEC;  // 512*1600
  float* c_st  = ex;                 ex += (size_t)BB * HDEC;
  float* h_st  = ex;                 ex += (size_t)BB * HDEC;
  float* hcat  = ex;                 ex += (size_t)BB * 2 * HENC;
  float* z_ws  = ex;                 ex += (size_t)BB * DLAT;
  float* zw[2] = { ex, ex + (size_t)BB * 4 * HDEC };
  ex += 2 * (size_t)BB * 4 * HDEC;
  float* d2buf = ex;                 /* NTOK*800 */
  float* encbuf = R0;   // NTOK*600, spans R0..R1 (e & hw temps dead by then)
  float* d1buf  = R0;   // NTOK*800, spans R0..R2 (enc & hw output dead)

  const float* NUL = nullptr;

  // ---- 1) embedding gather ----
  {
    int n = NTOK * DEMB;
    k_embed<<<(n + 255) / 256, 256, 0, stream>>>(x, emb, R0, n);
  }

  // ---- 2) two highway layers ----
  // hw(x): t1=relu(x@w1+b1); t2=relu(t1@w2+b2); t3=relu(x@w3+b3);
  //        t4=t3@w4+b4; out = sig(x@w5+b5)*t2 + (1-sig)*t4   (ACT=4 fused)
  {
    // layer 0: x=R0 -> out=R4
    gemm<1,false,false,true>(stream, R0, DEMB, hw0w[0], DEMB, DEMB, NUL,0,NUL,0,0, hw0b[0], NUL,NUL,NUL, R1, DEMB, NTOK, DEMB);
    gemm<1,false,false,true>(stream, R1, DEMB, hw0w[1], DEMB, DEMB, NUL,0,NUL,0,0, hw0b[1], NUL,NUL,NUL, R2, DEMB, NTOK, DEMB);
    gemm<1,false,false,true>(stream, R0, DEMB, hw0w[2], DEMB, DEMB, NUL,0,NUL,0,0, hw0b[2], NUL,NUL,NUL, R1, DEMB, NTOK, DEMB);
    gemm<0,false,false,true>(stream, R1, DEMB, hw0w[3], DEMB, DEMB, NUL,0,NUL,0,0, hw0b[3], NUL,NUL,NUL, R3, DEMB, NTOK, DEMB);
    gemm<4,false,false,true>(stream, R0, DEMB, hw0w[4], DEMB, DEMB, NUL,0,NUL,0,0, hw0b[4], NUL, R2, R3, R4, DEMB, NTOK, DEMB);
    // layer 1: x=R4 -> out=R2
    gemm<1,false,false,true>(stream, R4, DEMB, hw1w[0], DEMB, DEMB, NUL,0,NUL,0,0, hw1b[0], NUL,NUL,NUL, R0, DEMB, NTOK, DEMB);
    gemm<1,false,false,true>(stream, R0, DEMB, hw1w[1], DEMB, DEMB, NUL,0,NUL,0,0, hw1b[1], NUL,NUL,NUL, R1, DEMB, NTOK, DEMB);
    gemm<1,false,false,true>(stream, R4, DEMB, hw1w[2], DEMB, DEMB, NUL,0,NUL,0,0, hw1b[2], NUL,NUL,NUL, R0, DEMB, NTOK, DEMB);
    gemm<0,false,false,true>(stream, R0, DEMB, hw1w[3], DEMB, DEMB, NUL,0,NUL,0,0, hw1b[3], NUL,NUL,NUL, R3, DEMB, NTOK, DEMB);
    gemm<4,false,false,true>(stream, R4, DEMB, hw1w[4], DEMB, DEMB, NUL,0,NUL,0,0, hw1b[4], NUL, R1, R3, R2, DEMB, NTOK, DEMB);
  }
  float* hbuf = R2;   // highway output (NTOK x 300), alive through encoder

  // ---- 3) bidirectional encoder LSTM (H=300) ----
  // per step: gates = x_t@Wi + h@Wh + b  (dual-slot GEMM), then cell.
  for (int dir = 0; dir < 2; ++dir) {
    fill0(stream, c_st, BB * HENC);
    fill0(stream, h_st, BB * HENC);
    for (int s = 0; s < SEQL; ++s) {
      int t = dir ? (SEQL - 1 - s) : s;
      gemm<0,true,false,true>(stream,
           hbuf + (size_t)t * DEMB, SEQL * DEMB, enc_wi[dir], 4 * HENC, DEMB,
           h_st, HENC,               enc_wh[dir], 4 * HENC, HENC,
           enc_bb[dir], NUL, NUL, NUL, gates, 4 * HENC, BB, 4 * HENC);
      int n = BB * HENC;
      k_cell<<<(n + 255) / 256, 256, 0, stream>>>(gates, c_st, h_st, HENC,
                                                  encbuf, 2 * HENC, dir ? HENC : 0, t);
    }
  }

  // ---- 4) hcat, mean, logvar, z ----
  {
    int n = BB * 2 * HENC;
    k_hcat<<<(n + 255) / 256, 256, 0, stream>>>(encbuf, hcat);
    gemm<3,false,false,true>(stream, hcat, 2 * HENC, mean_w, DLAT, 2 * HENC, NUL,0,NUL,0,0,
         mean_b, NUL, NUL, NUL, out_mean, DLAT, BB, DLAT);     // 4*tanh
    gemm<0,false,false,true>(stream, hcat, 2 * HENC, logv_w, DLAT, 2 * HENC, NUL,0,NUL,0,0,
         logv_b, NUL, NUL, NUL, out_logvar, DLAT, BB, DLAT);
    int m = BB * DLAT;
    k_z<<<(m + 255) / 256, 256, 0, stream>>>(out_mean, out_logvar, eps, out_z, z_ws);
  }

  // ---- 5) decoder layer 1 (z constant over t: precompute z@Wi+b once) ----
  for (int dir = 0; dir < 2; ++dir) {
    gemm<0,false,false,true>(stream, z_ws, DLAT, d1_wi[dir], 4 * HDEC, DLAT, NUL,0,NUL,0,0,
         d1_bb[dir], NUL, NUL, NUL, zw[dir], 4 * HDEC, BB, 4 * HDEC);
    fill0(stream, c_st, BB * HDEC);
    fill0(stream, h_st, BB * HDEC);
    for (int s = 0; s < SEQL; ++s) {
      int t = dir ? (SEQL - 1 - s) : s;
      gemm<0,false,true,false>(stream,
           h_st, HDEC, d1_wh[dir], 4 * HDEC, HDEC, NUL,0,NUL,0,0,
           NUL, zw[dir], NUL, NUL, gates, 4 * HDEC, BB, 4 * HDEC);
      int n = BB * HDEC;
      k_cell<<<(n + 255) / 256, 256, 0, stream>>>(gates, c_st, h_st, HDEC,
                                                  d1buf, 2 * HDEC, dir ? HDEC : 0, t);
    }
  }

  // ---- 6) decoder layer 2 (dual-slot: d1_t@Wi + h@Wh + b) ----
  for (int dir = 0; dir < 2; ++dir) {
    fill0(stream, c_st, BB * HDEC);
    fill0(stream, h_st, BB * HDEC);
    for (int s = 0; s < SEQL; ++s) {
      int t = dir ? (SEQL - 1 - s) : s;
      gemm<0,true,false,true>(stream,
           d1buf + (size_t)t * 2 * HDEC, SEQL * 2 * HDEC, d2_wi[dir], 4 * HDEC, 2 * HDEC,
           h_st, HDEC,                                    d2_wh[dir], 4 * HDEC, HDEC,
           d2_bb[dir], NUL, NUL, NUL, gates, 4 * HDEC, BB, 4 * HDEC);
      int n = BB * HDEC;
      k_cell<<<(n + 255) / 256, 256, 0, stream>>>(gates, c_st, h_st, HDEC,
                                                  d2buf, 2 * HDEC, dir ? HDEC : 0, t);
    }
  }

  // ---- 7) vocab projection ----
  gemm<0,false,false,true>(stream, d2buf, 2 * HDEC, out_w, NVOCAB, 2 * HDEC, NUL,0,NUL,0,0,
       out_b, NUL, NUL, NUL, out_logits, NVOCAB, NTOK, NVOCAB);
}